// DNonlocal_48507360641765
// MI455X (gfx1250) — compile-verified
//
#include <hip/hip_runtime.h>
#include <math.h>

typedef float v2f __attribute__((ext_vector_type(2)));
typedef float v4f __attribute__((ext_vector_type(4)));
typedef float v8f __attribute__((ext_vector_type(8)));

#define NB    16
#define CCH   512
#define HH    64
#define WW    64
#define HWF   4096   // 64*64
#define RHW   1024   // 32*32 reduced plane
#define LENI  256
#define MPAD  272    // 17*16 padded I-rows (only 0..256 semantically used)
#define NHWC  65536  // N*H*W per channel for BN
#define BN_EPS 1e-5f

// ---------------------------------------------------------------------------
// Pack l[:, :, ::2, ::2] contiguous: lrp[n][c][k], k in [0,1024). Fixes the
// stride-2 load pattern once, enabling b64 vector loads in the GEMMs.
// ---------------------------------------------------------------------------
__global__ void k_pack(const float* __restrict__ l, float* __restrict__ lrp) {
    const size_t idx = (size_t)blockIdx.x * blockDim.x + threadIdx.x;
    const size_t total = (size_t)NB * CCH * RHW;
    if (idx >= total) return;
    const int k = (int)(idx & (RHW - 1));
    const size_t plane = idx >> 10;          // n*CCH + c
    lrp[idx] = l[plane * HWF + ((k >> 5) << 7) + ((k & 31) << 1)];
}

// ---------------------------------------------------------------------------
// Kernel A: f1[i,j] = sum_k phi[i,k]*g[j,k]  (M=272pad, N=512, K=1024) per n
// phi[i,k] = (1-sd)*lrp[i,k] + sd*lrp[i+1,k] ; g[j,k] = lrp[j,k]
// Each wave: one 16-row stripe x TWO 16-col tiles (shared A fragment).
// Output stored transposed: f1T[n][j][i] (i contiguous -> v4f stores).
// ---------------------------------------------------------------------------
__global__ void k_f1_gemm(const float* __restrict__ lrp, const float* __restrict__ sdp,
                          float* __restrict__ f1T) {
    const int wave = threadIdx.x >> 5;
    const int lane = threadIdx.x & 31;
    const int TOT = NB * (MPAD / 16) * (CCH / 32);         // 16*17*16 = 4352
    const int tile = blockIdx.x * (blockDim.x >> 5) + wave;
    if (tile >= TOT) return;
    const int n = tile / ((MPAD / 16) * (CCH / 32));
    int r = tile % ((MPAD / 16) * (CCH / 32));
    const int i0 = (r / (CCH / 32)) * 16;
    const int j0 = (r % (CCH / 32)) * 32;
    const int half = lane >> 4;                            // K pair select
    const int mr   = lane & 15;
    const float sd = sdp[0], csd = 1.0f - sd;

    const float* ra  = lrp + ((size_t)n * CCH + i0 + mr) * RHW;        // row i
    const float* ra1 = ra + RHW;                                       // row i+1
    const float* rb0 = lrp + ((size_t)n * CCH + j0 + mr) * RHW;        // col j0+mr
    const float* rb1 = rb0 + (size_t)16 * RHW;                         // col j0+16+mr

    v8f acc0 = {}, acc1 = {};
    for (int k0 = 0; k0 < RHW; k0 += 4) {
        const int ka = k0 + (half << 1);
        const v2f x  = *(const v2f*)(ra  + ka);
        const v2f x1 = *(const v2f*)(ra1 + ka);
        v2f a;
        a.x = csd * x.x + sd * x1.x;
        a.y = csd * x.y + sd * x1.y;
        const v2f b0 = *(const v2f*)(rb0 + ka);
        const v2f b1 = *(const v2f*)(rb1 + ka);
        acc0 = __builtin_amdgcn_wmma_f32_16x16x4_f32(false, a, false, b0, (short)0, acc0,
                                                     false, false);
        acc1 = __builtin_amdgcn_wmma_f32_16x16x4_f32(false, a, false, b1, (short)0, acc1,
                                                     false, false);
    }
    // C-fragment: lane holds (i = i0+v+8*half, j = jcol); i contiguous over v.
    float* o0 = f1T + ((size_t)n * CCH + j0 + mr) * MPAD + i0 + (half << 3);
    float* o1 = o0 + (size_t)16 * MPAD;
    *(v4f*)(o0)     = (v4f){acc0[0], acc0[1], acc0[2], acc0[3]};
    *(v4f*)(o0 + 4) = (v4f){acc0[4], acc0[5], acc0[6], acc0[7]};
    *(v4f*)(o1)     = (v4f){acc1[0], acc1[1], acc1[2], acc1[3]};
    *(v4f*)(o1 + 4) = (v4f){acc1[4], acc1[5], acc1[6], acc1[7]};
}

// ---------------------------------------------------------------------------
// Kernel B: f2[j,hw] = sum_{i<256} f1T[j,i]*theta[i,hw] (WMMA, K=256) per n,
// + i=256 rank-1 correction, fused dual-floor length interp:
//   f = floor(f2/64)*64*(1-ld) + floor((f2+extra)/64)*64*ld
// Each wave: one 16-row (j) stripe x TWO 16-col (hw) tiles, shared A fragment.
// ---------------------------------------------------------------------------
__global__ void k_f2_gemm(const float* __restrict__ l, const float* __restrict__ f1T,
                          const float* __restrict__ sdp, const float* __restrict__ ldp,
                          float* __restrict__ f) {
    const int wave = threadIdx.x >> 5;
    const int lane = threadIdx.x & 31;
    const int TOT = NB * (CCH / 16) * (HWF / 32);          // 16*32*128 = 65536
    const int tile = blockIdx.x * (blockDim.x >> 5) + wave;
    if (tile >= TOT) return;
    const int n = tile / ((CCH / 16) * (HWF / 32));
    int r = tile % ((CCH / 16) * (HWF / 32));
    const int j0  = (r / (HWF / 32)) * 16;
    const int hw0 = (r % (HWF / 32)) * 32;
    const int half = lane >> 4;
    const int mr   = lane & 15;
    const float sd = sdp[0], csd = 1.0f - sd;
    const float ld = ldp[0], cld = 1.0f - ld;

    const float* lp = l + (size_t)n * CCH * HWF;           // theta source rows
    const float* ar = f1T + ((size_t)n * CCH + j0 + mr) * MPAD;   // A row j0+mr
    const int col0 = hw0 + mr;
    const int col1 = col0 + 16;

    v8f acc0 = {}, acc1 = {};
    for (int k0 = 0; k0 < LENI; k0 += 4) {
        const int ka = k0 + (half << 1);
        const v2f a = *(const v2f*)(ar + ka);              // f1T[j, ka..ka+1]
        __builtin_prefetch(lp + (size_t)(ka + 16) * HWF + col0, 0, 1);
        const float t0a = lp[(size_t)ka * HWF + col0];
        const float t1a = lp[(size_t)(ka + 1) * HWF + col0];
        const float t2a = lp[(size_t)(ka + 2) * HWF + col0];
        const float t0b = lp[(size_t)ka * HWF + col1];
        const float t1b = lp[(size_t)(ka + 1) * HWF + col1];
        const float t2b = lp[(size_t)(ka + 2) * HWF + col1];
        v2f b0, b1;
        b0.x = csd * t0a + sd * t1a;  b0.y = csd * t1a + sd * t2a;
        b1.x = csd * t0b + sd * t1b;  b1.y = csd * t1b + sd * t2b;
        acc0 = __builtin_amdgcn_wmma_f32_16x16x4_f32(false, a, false, b0, (short)0, acc0,
                                                     false, false);
        acc1 = __builtin_amdgcn_wmma_f32_16x16x4_f32(false, a, false, b1, (short)0, acc1,
                                                     false, false);
    }
    const float th0 = csd * lp[(size_t)256 * HWF + col0] + sd * lp[(size_t)257 * HWF + col0];
    const float th1 = csd * lp[(size_t)256 * HWF + col1] + sd * lp[(size_t)257 * HWF + col1];
    float* fout = f + (size_t)n * CCH * HWF;
    #pragma unroll
    for (int v = 0; v < 8; ++v) {
        const int j = j0 + v + (half << 3);
        const float f1_256 = f1T[((size_t)n * CCH + j) * MPAD + 256];
        const float b0 = acc0[v], e0 = f1_256 * th0;
        const float b1 = acc1[v], e1 = f1_256 * th1;
        const float fs0 = floorf(b0 * (1.0f / 64.0f)) * 64.0f;
        const float fl0 = floorf((b0 + e0) * (1.0f / 64.0f)) * 64.0f;
        const float fs1 = floorf(b1 * (1.0f / 64.0f)) * 64.0f;
        const float fl1 = floorf((b1 + e1) * (1.0f / 64.0f)) * 64.0f;
        fout[(size_t)j * HWF + col0] = fs0 * cld + fl0 * ld;
        fout[(size_t)j * HWF + col1] = fs1 * cld + fl1 * ld;
    }
}

// ---------------------------------------------------------------------------
// Kernel C: depthwise 3x3 conv stats ONLY (no y materialization).
// One block per (n, c) plane; LDS tree reduce + one atomicAdd pair per block.
// ---------------------------------------------------------------------------
__global__ void k_conv_stats(const float* __restrict__ f, const float* __restrict__ cw,
                             float* __restrict__ stats) {
    __shared__ float s1[256];
    __shared__ float s2[256];
    const int plane = blockIdx.x;            // n*CCH + c
    const int c = plane & (CCH - 1);
    float w[9];
    #pragma unroll
    for (int t = 0; t < 9; ++t) w[t] = cw[c * 9 + t];
    const float* fp = f + (size_t)plane * HWF;
    float lsum = 0.0f, lsq = 0.0f;
    for (int p = threadIdx.x; p < HWF; p += blockDim.x) {
        const int h = p >> 6, x = p & 63;
        float acc = 0.0f;
        #pragma unroll
        for (int dy = -1; dy <= 1; ++dy) {
            const int hh = h + dy;
            if (hh < 0 || hh >= HH) continue;
            #pragma unroll
            for (int dx = -1; dx <= 1; ++dx) {
                const int xx = x + dx;
                if (xx < 0 || xx >= WW) continue;
                acc += w[(dy + 1) * 3 + (dx + 1)] * fp[hh * WW + xx];
            }
        }
        lsum += acc;
        lsq  += acc * acc;
    }
    s1[threadIdx.x] = lsum;
    s2[threadIdx.x] = lsq;
    __syncthreads();
    for (int off = blockDim.x >> 1; off > 0; off >>= 1) {
        if (threadIdx.x < (unsigned)off) {
            s1[threadIdx.x] += s1[threadIdx.x + off];
            s2[threadIdx.x] += s2[threadIdx.x + off];
        }
        __syncthreads();
    }
    if (threadIdx.x == 0) {
        atomicAdd(&stats[c], s1[0]);
        atomicAdd(&stats[CCH + c], s2[0]);
    }
}

// ---------------------------------------------------------------------------
// Kernel E: recompute conv (f is hot in L2; 9x spatial reuse) fused with
// batchnorm (population var) + affine + residual -> d_out. Saves the y
// write+read round-trip (~268 MB HBM).
// ---------------------------------------------------------------------------
__global__ void k_conv_bn_res(const float* __restrict__ f, const float* __restrict__ cw,
                              const float* __restrict__ l, const float* __restrict__ gamma,
                              const float* __restrict__ beta, const float* __restrict__ stats,
                              float* __restrict__ out) {
    const size_t idx = (size_t)blockIdx.x * blockDim.x + threadIdx.x;
    const size_t total = (size_t)NB * CCH * HWF;
    if (idx >= total) return;
    const size_t plane = idx >> 12;          // n*CCH + c (uniform per 256-block)
    const int c = (int)(plane & (CCH - 1));
    const int p = (int)(idx & (HWF - 1));
    const int h = p >> 6, x = p & 63;
    const float* fp = f + plane * HWF;
    float acc = 0.0f;
    #pragma unroll
    for (int dy = -1; dy <= 1; ++dy) {
        const int hh = h + dy;
        if (hh < 0 || hh >= HH) continue;
        #pragma unroll
        for (int dx = -1; dx <= 1; ++dx) {
            const int xx = x + dx;
            if (xx < 0 || xx >= WW) continue;
            acc += cw[c * 9 + (dy + 1) * 3 + (dx + 1)] * fp[hh * WW + xx];
        }
    }
    const float mean = stats[c] * (1.0f / NHWC);
    const float var  = stats[CCH + c] * (1.0f / NHWC) - mean * mean;
    const float sc   = rsqrtf(var + BN_EPS) * gamma[c];
    out[idx] = (acc - mean) * sc + beta[c] + l[idx];
}

__global__ void k_zero(float* __restrict__ p, int n) {
    int i = blockIdx.x * blockDim.x + threadIdx.x;
    if (i < n) p[i] = 0.0f;
}

extern "C" void kernel_launch(void* const* d_in, const int* in_sizes, int n_in,
                              void* d_out, int out_size, void* d_ws, size_t ws_size,
                              hipStream_t stream) {
    (void)in_sizes; (void)n_in; (void)out_size; (void)ws_size;
    const float* l     = (const float*)d_in[0];
    const float* cw    = (const float*)d_in[1];
    const float* gamma = (const float*)d_in[2];
    const float* beta  = (const float*)d_in[3];
    const float* sdp   = (const float*)d_in[4];
    const float* ldp   = (const float*)d_in[5];
    float* out = (float*)d_out;
    float* ws  = (float*)d_ws;

    float* lrp   = ws;                                     // 16*512*1024  (~33.5 MB)
    float* f1T   = lrp + (size_t)NB * CCH * RHW;           // 16*512*272   (~8.9 MB)
    float* stats = f1T + (size_t)NB * CCH * MPAD;          // 2*512
    float* f     = stats + 2 * CCH;                        // 16*512*4096  (~134 MB)

    k_zero<<<4, 256, 0, stream>>>(stats, 2 * CCH);
    {
        const size_t total = (size_t)NB * CCH * RHW;
        k_pack<<<(unsigned)((total + 255) / 256), 256, 0, stream>>>(l, lrp);
    }
    {
        const int tiles = NB * (MPAD / 16) * (CCH / 32);   // 4352 waves
        k_f1_gemm<<<(tiles + 7) / 8, 256, 0, stream>>>(lrp, sdp, f1T);
    }
    {
        const int tiles = NB * (CCH / 16) * (HWF / 32);    // 65536 waves
        k_f2_gemm<<<(tiles + 7) / 8, 256, 0, stream>>>(l, f1T, sdp, ldp, f);
    }
    k_conv_stats<<<NB * CCH, 256, 0, stream>>>(f, cw, stats);
    {
        const size_t total = (size_t)NB * CCH * HWF;
        k_conv_bn_res<<<(unsigned)((total + 255) / 256), 256, 0, stream>>>(
            f, cw, l, gamma, beta, stats, out);
    }
}